// GCNClassifier_23845658428048
// MI455X (gfx1250) — compile-verified
//
#include <hip/hip_runtime.h>
#include <hip/hip_bf16.h>

// ---------------------------------------------------------------------------
// GCN classifier for MI455X (gfx1250, wave32).
//   h1 = relu(graphnorm(x @ W1 + b1))          -- WMMA f32 16x16x4
//   h2 = relu(graphnorm(h1 @ W2 + b2))         -- WMMA f32 16x16x4
//   p  = APPNP(h2, edges, K=20, alpha=0.9)     -- L2-resident gather/scatter
//   out = p @ Wout + bout                      -- small VALU gemm
// ---------------------------------------------------------------------------

#define NNODES 100000
#define CH     128
#define NCLS   10
#define NEDGE  3200000
#define ALPHA  0.9f
#define KPROP  20
#define GNEPS  1e-5f

typedef float v2f __attribute__((ext_vector_type(2)));
typedef float v8f __attribute__((ext_vector_type(8)));

// ---------------------------------------------------------------------------
// Dense GEMM: Y[N x 128] = X[N x 128] @ W[128 x 128] + bias, via
// V_WMMA_F32_16X16X4_F32.  One block = 16 rows x 128 cols; 8 waves, each wave
// owns one 16x16 output tile; K-loop of 32 WMMA steps.
// ---------------------------------------------------------------------------
__global__ void __launch_bounds__(256)
gemm128_wmma_kernel(const float* __restrict__ X, const float* __restrict__ W,
                    const float* __restrict__ bias, float* __restrict__ Y)
{
    const int wave = threadIdx.x >> 5;          // 0..7  -> column tile
    const int lane = threadIdx.x & 31;
    const int m    = lane & 15;                 // A row / B,D column-in-tile
    const int hi   = lane >> 4;                 // lane half selects K pair
    const long long row0 = (long long)blockIdx.x * 16;
    const int col0 = wave * 16;

    const float* __restrict__ arow = X + (row0 + m) * CH;

    v8f acc = {};
#pragma unroll
    for (int kb = 0; kb < CH; kb += 4) {
        const int k0 = kb + 2 * hi;
        v2f av; av.x = arow[k0];              av.y = arow[k0 + 1];
        v2f bv; bv.x = W[(long long)k0 * CH + col0 + m];
                bv.y = W[(long long)(k0 + 1) * CH + col0 + m];
        acc = __builtin_amdgcn_wmma_f32_16x16x4_f32(
            /*neg_a=*/false, av, /*neg_b=*/false, bv,
            /*c_mod=*/(short)0, acc, /*reuse_a=*/false, /*reuse_b=*/false);
    }

    const float bc = bias[col0 + m];
#pragma unroll
    for (int v = 0; v < 8; ++v) {
        const int row = v + 8 * hi;           // f32 C/D VGPR layout
        Y[(row0 + row) * CH + col0 + m] = acc[v] + bc;
    }
}

// ---------------------------------------------------------------------------
// GraphNorm statistics: per-channel sum and sum-of-squares over all nodes.
// ---------------------------------------------------------------------------
__global__ void zero_stats_kernel(float* __restrict__ ssum, float* __restrict__ ssq)
{
    const int c = threadIdx.x;
    if (c < CH) { ssum[c] = 0.0f; ssq[c] = 0.0f; }
}

__global__ void __launch_bounds__(128)
colstats_kernel(const float* __restrict__ H, float* __restrict__ ssum,
                float* __restrict__ ssq)
{
    const int c = threadIdx.x;                  // channel
    float s = 0.0f, q = 0.0f;
    for (long long r = blockIdx.x; r < NNODES; r += gridDim.x) {
        const float v = H[r * CH + c];
        s += v; q += v * v;
    }
    unsafeAtomicAdd(&ssum[c], s);               // global_atomic_add_f32
    unsafeAtomicAdd(&ssq[c],  q);
}

// a[c] = w[c]/sqrt(var+eps), sh[c] = mean*mean_scale
__global__ void finstats_kernel(const float* __restrict__ ssum,
                                const float* __restrict__ ssq,
                                const float* __restrict__ gw,
                                const float* __restrict__ gms,
                                float* __restrict__ ga, float* __restrict__ gsh)
{
    const int c = threadIdx.x;
    if (c >= CH) return;
    const float invN = 1.0f / (float)NNODES;
    const float mean = ssum[c] * invN;
    const float ex2  = ssq[c] * invN;
    const float mms  = mean * gms[c];
    const float var  = ex2 - 2.0f * mean * mms + mms * mms;
    ga[c]  = gw[c] * rsqrtf(var + GNEPS);
    gsh[c] = mms;
}

__global__ void __launch_bounds__(256)
norm_relu_kernel(float* __restrict__ H, const float* __restrict__ ga,
                 const float* __restrict__ gsh, const float* __restrict__ gb)
{
    const long long idx = (long long)blockIdx.x * blockDim.x + threadIdx.x;
    if (idx >= (long long)NNODES * CH) return;
    const int c = (int)(idx & (CH - 1));
    const float y = ga[c] * (H[idx] - gsh[c]) + gb[c];
    H[idx] = fmaxf(y, 0.0f);
}

// ---------------------------------------------------------------------------
// Edge preprocessing: int64 -> int32, in-degree (incl. self loop), gcn norm.
// ---------------------------------------------------------------------------
__global__ void deg_init_kernel(unsigned* __restrict__ deg)
{
    const long long i = (long long)blockIdx.x * blockDim.x + threadIdx.x;
    if (i < NNODES) deg[i] = 1u;                // self loop
}

__global__ void prep_edges_kernel(const long long* __restrict__ ei,
                                  int* __restrict__ esrc, int* __restrict__ edst,
                                  unsigned* __restrict__ deg)
{
    const long long e = (long long)blockIdx.x * blockDim.x + threadIdx.x;
    if (e >= NEDGE) return;
    const int s = (int)ei[e];
    const int d = (int)ei[(long long)NEDGE + e];
    esrc[e] = s; edst[e] = d;
    atomicAdd(&deg[d], 1u);
}

__global__ void node_norm_kernel(const unsigned* __restrict__ deg,
                                 float* __restrict__ dinv,
                                 float* __restrict__ selfc)
{
    const long long i = (long long)blockIdx.x * blockDim.x + threadIdx.x;
    if (i >= NNODES) return;
    const float di = rsqrtf((float)deg[i]);
    dinv[i]  = di;
    selfc[i] = (1.0f - ALPHA) * di * di;        // (1-a) * self-loop norm
}

__global__ void edge_norm_kernel(const int* __restrict__ esrc,
                                 const int* __restrict__ edst,
                                 const float* __restrict__ dinv,
                                 float* __restrict__ enorm)
{
    const long long e = (long long)blockIdx.x * blockDim.x + threadIdx.x;
    if (e >= NEDGE) return;
    enorm[e] = (1.0f - ALPHA) * dinv[esrc[e]] * dinv[edst[e]];
}

// ---------------------------------------------------------------------------
// APPNP iteration:  xout = alpha*h + (1-a)*selfnorm*xin  (init)
//                   xout[dst] += (1-a)*norm*xin[src]     (edges, fp atomics)
// One wave per edge: 32 lanes x float4 = one coalesced 512B row.
// ---------------------------------------------------------------------------
__global__ void __launch_bounds__(256)
appnp_init_kernel(const float* __restrict__ h, const float* __restrict__ xin,
                  const float* __restrict__ selfc, float* __restrict__ xout)
{
    const long long idx = (long long)blockIdx.x * blockDim.x + threadIdx.x;
    if (idx >= (long long)NNODES * CH) return;
    const long long node = idx >> 7;
    xout[idx] = ALPHA * h[idx] + selfc[node] * xin[idx];
}

__global__ void __launch_bounds__(256)
appnp_edges_kernel(const int* __restrict__ esrc, const int* __restrict__ edst,
                   const float* __restrict__ enorm,
                   const float* __restrict__ xin, float* __restrict__ xout)
{
    const long long t = (long long)blockIdx.x * blockDim.x + threadIdx.x;
    const long long e = t >> 5;                 // one wave per edge
    if (e >= NEDGE) return;
    const int ch = (int)(t & 31) * 4;           // lane -> 4 contiguous floats
    const int s = esrc[e];
    const int d = edst[e];
    const float w = enorm[e];
    const float4 v = *(const float4*)(xin + (long long)s * CH + ch);
    float* o = xout + (long long)d * CH + ch;
    unsafeAtomicAdd(o + 0, w * v.x);
    unsafeAtomicAdd(o + 1, w * v.y);
    unsafeAtomicAdd(o + 2, w * v.z);
    unsafeAtomicAdd(o + 3, w * v.w);
}

// ---------------------------------------------------------------------------
// Output projection: [N x 128] @ [128 x 10] + bout.
// ---------------------------------------------------------------------------
__global__ void __launch_bounds__(256)
gemm_out_kernel(const float* __restrict__ X, const float* __restrict__ W,
                const float* __restrict__ b, float* __restrict__ Y)
{
    const long long idx = (long long)blockIdx.x * blockDim.x + threadIdx.x;
    if (idx >= (long long)NNODES * NCLS) return;
    const long long node = idx / NCLS;
    const int c = (int)(idx % NCLS);
    const float* __restrict__ xr = X + node * CH;
    float acc = b[c];
#pragma unroll 8
    for (int k = 0; k < CH; ++k) acc += xr[k] * W[(long long)k * NCLS + c];
    Y[idx] = acc;
}

// ---------------------------------------------------------------------------
extern "C" void kernel_launch(void* const* d_in, const int* in_sizes, int n_in,
                              void* d_out, int out_size, void* d_ws, size_t ws_size,
                              hipStream_t stream)
{
    const float*     x    = (const float*)d_in[0];
    const long long* ei   = (const long long*)d_in[1];
    const float*     W1   = (const float*)d_in[2];
    const float*     b1   = (const float*)d_in[3];
    const float*     g1w  = (const float*)d_in[4];
    const float*     g1b  = (const float*)d_in[5];
    const float*     g1ms = (const float*)d_in[6];
    const float*     W2   = (const float*)d_in[7];
    const float*     b2   = (const float*)d_in[8];
    const float*     g2w  = (const float*)d_in[9];
    const float*     g2b  = (const float*)d_in[10];
    const float*     g2ms = (const float*)d_in[11];
    const float*     Wout = (const float*)d_in[12];
    const float*     bout = (const float*)d_in[13];
    float*           out  = (float*)d_out;

    // ---- workspace carve (256B aligned) -----------------------------------
    char* base = (char*)d_ws;
    size_t off = 0;
    auto carve = [&](size_t bytes) -> char* {
        char* p = base + off;
        off = (off + bytes + 255) & ~(size_t)255;
        return p;
    };
    const size_t NC = (size_t)NNODES * CH;
    float*    H1    = (float*)carve(NC * sizeof(float));      // layer1 / ping
    float*    H2    = (float*)carve(NC * sizeof(float));      // layer2 = h
    float*    P     = (float*)carve(NC * sizeof(float));      // pong
    unsigned* deg   = (unsigned*)carve(NNODES * sizeof(unsigned));
    float*    dinv  = (float*)carve(NNODES * sizeof(float));
    float*    selfc = (float*)carve(NNODES * sizeof(float));
    int*      esrc  = (int*)carve(NEDGE * sizeof(int));
    int*      edst  = (int*)carve(NEDGE * sizeof(int));
    float*    enorm = (float*)carve(NEDGE * sizeof(float));
    float*    ssum  = (float*)carve(CH * sizeof(float));
    float*    ssq   = (float*)carve(CH * sizeof(float));
    float*    ga    = (float*)carve(CH * sizeof(float));
    float*    gsh   = (float*)carve(CH * sizeof(float));
    (void)ws_size; (void)n_in; (void)in_sizes; (void)out_size;

    const int nodeBlocks = (NNODES + 255) / 256;
    const int edgeBlocks = (NEDGE + 255) / 256;
    const int elemBlocks = (int)((NC + 255) / 256);
    const int rowTiles   = NNODES / 16;                        // 6250

    // ---- edge preprocessing (once per launch, deterministic) --------------
    deg_init_kernel<<<nodeBlocks, 256, 0, stream>>>(deg);
    prep_edges_kernel<<<edgeBlocks, 256, 0, stream>>>(ei, esrc, edst, deg);
    node_norm_kernel<<<nodeBlocks, 256, 0, stream>>>(deg, dinv, selfc);
    edge_norm_kernel<<<edgeBlocks, 256, 0, stream>>>(esrc, edst, dinv, enorm);

    // ---- layer 1: GEMM + GraphNorm + ReLU ---------------------------------
    gemm128_wmma_kernel<<<rowTiles, 256, 0, stream>>>(x, W1, b1, H1);
    zero_stats_kernel<<<1, 128, 0, stream>>>(ssum, ssq);
    colstats_kernel<<<512, 128, 0, stream>>>(H1, ssum, ssq);
    finstats_kernel<<<1, 128, 0, stream>>>(ssum, ssq, g1w, g1ms, ga, gsh);
    norm_relu_kernel<<<elemBlocks, 256, 0, stream>>>(H1, ga, gsh, g1b);

    // ---- layer 2: GEMM + GraphNorm + ReLU ---------------------------------
    gemm128_wmma_kernel<<<rowTiles, 256, 0, stream>>>(H1, W2, b2, H2);
    zero_stats_kernel<<<1, 128, 0, stream>>>(ssum, ssq);
    colstats_kernel<<<512, 128, 0, stream>>>(H2, ssum, ssq);
    finstats_kernel<<<1, 128, 0, stream>>>(ssum, ssq, g2w, g2ms, ga, gsh);
    norm_relu_kernel<<<elemBlocks, 256, 0, stream>>>(H2, ga, gsh, g2b);

    // ---- APPNP: K=20 propagation steps, L2-resident ping-pong -------------
    const long long edgeThreads = (long long)NEDGE * 32;
    const int edgeGrid = (int)((edgeThreads + 255) / 256);
    const float* xin = H2;                       // k = 0 reads h directly
    for (int k = 0; k < KPROP; ++k) {
        float* xout = (k & 1) ? P : H1;
        appnp_init_kernel<<<elemBlocks, 256, 0, stream>>>(H2, xin, selfc, xout);
        appnp_edges_kernel<<<edgeGrid, 256, 0, stream>>>(esrc, edst, enorm, xin, xout);
        xin = xout;
    }
    // KPROP = 20 (even) -> final result lives in P

    // ---- output projection -------------------------------------------------
    const int outBlocks = (int)(((long long)NNODES * NCLS + 255) / 256);
    gemm_out_kernel<<<outBlocks, 256, 0, stream>>>(P, Wout, bout, out);
}